// GraphConv_25056839204918
// MI455X (gfx1250) — compile-verified
//
#include <hip/hip_runtime.h>
#include <math.h>

typedef __attribute__((ext_vector_type(2))) float v2f;
typedef __attribute__((ext_vector_type(8))) float v8f;

#define DD 64

// ---- CDNA5 async global->LDS copy helpers (ASYNCcnt-tracked, no VGPR hop) ----
// LDS byte offset = low 32 bits of the generic pointer to a __shared__ object
// (shared->flat addrspacecast: high dword = LDS aperture, low dword = offset).
__device__ __forceinline__ void async_copy_16k(void* lds_dst, const void* gsrc, int tid) {
    unsigned ldsa = (unsigned)(size_t)lds_dst + (unsigned)tid * 16u;
    unsigned voff = (unsigned)tid * 16u;
#pragma unroll
    for (int j = 0; j < 8; ++j) {   // 128 threads * 8 * 16B = 16KB
        asm volatile("global_load_async_to_lds_b128 %0, %1, %2"
                     :: "v"(ldsa + (unsigned)j * 2048u),
                        "v"(voff + (unsigned)j * 2048u),
                        "s"(gsrc)
                     : "memory");
    }
}
__device__ __forceinline__ void async_wait_all() {
    asm volatile("s_wait_asynccnt 0x0" ::: "memory");
}

// out(16x64) = A(16x64) * W(64x64), all row-major in LDS, one wave32.
// V_WMMA_F32_16X16X4_F32: A frag = 2 VGPR (lanes0-15:K0,K1; lanes16-31:K2,K3),
// B frag mirrored, C/D = 8 VGPR (lane half selects M / M+8).
__device__ __forceinline__ void tile_gemm_16x64(const float* __restrict__ sA,
                                                const float* __restrict__ sW,
                                                float* __restrict__ sO, int lane) {
    const int half = lane >> 4;
    const int mn   = lane & 15;
#pragma unroll
    for (int nt = 0; nt < 4; ++nt) {
        v8f acc = {};
#pragma unroll
        for (int kk = 0; kk < 16; ++kk) {
            const int kb = kk * 4 + half * 2;
            v2f a, b;
            a.x = sA[mn * DD + kb];
            a.y = sA[mn * DD + kb + 1];
            b.x = sW[kb * DD + nt * 16 + mn];
            b.y = sW[(kb + 1) * DD + nt * 16 + mn];
            acc = __builtin_amdgcn_wmma_f32_16x16x4_f32(
                false, a, false, b, (short)0, acc, false, false);
        }
#pragma unroll
        for (int r = 0; r < 8; ++r)
            sO[(r + half * 8) * DD + nt * 16 + mn] = acc[r];
    }
}

// ---------------- Kernel 1: Qn = emb @ Wq  (per-node projection) --------------
__global__ void __launch_bounds__(128)
node_gemm_kernel(const float* __restrict__ emb, const float* __restrict__ W,
                 float* __restrict__ out, int N) {
    __shared__ float sW[DD * DD];
    __shared__ float sA[4][16 * DD];
    __shared__ float sO[4][16 * DD];
    const int tid = threadIdx.x, wave = tid >> 5, lane = tid & 31;

    async_copy_16k(sW, W, tid);                     // 16KB weight matrix

    const int rowBase = blockIdx.x * 64 + wave * 16;
    {   // async-stage the 16x64 A tile (4KB per wave), row-clamped at the tail
        unsigned ldsA = (unsigned)(size_t)(void*)sA[wave];
#pragma unroll
        for (int j = 0; j < 8; ++j) {
            int bi = j * 512 + lane * 16;           // byte index within tile
            int r = bi >> 8;
            int row = rowBase + r; if (row >= N) row = N - 1;
            unsigned voff = (unsigned)row * 256u + (unsigned)(bi & 255);
            asm volatile("global_load_async_to_lds_b128 %0, %1, %2"
                         :: "v"(ldsA + (unsigned)bi), "v"(voff), "s"(emb)
                         : "memory");
        }
    }
    async_wait_all();
    __syncthreads();
    tile_gemm_16x64(sA[wave], sW, sO[wave], lane);
    __syncthreads();
    for (int i = lane; i < 16 * DD / 4; i += 32) {
        int r = i >> 4, c4 = i & 15;
        int row = rowBase + r;
        if (row < N)
            ((float4*)(out + (size_t)row * DD))[c4] = ((float4*)sO[wave])[i];
    }
}

// ------- Kernel 2: per-edge K projection + attention logits + softmax denom ---
__global__ void __launch_bounds__(128)
edge_att_kernel(const float* __restrict__ emb, const float* __restrict__ relW,
                const float* __restrict__ Wk, const float* __restrict__ Qn,
                const int* __restrict__ eidx, const int* __restrict__ etype,
                float* __restrict__ exp_att, float* __restrict__ att_norm, int E) {
    __shared__ float sW[DD * DD];
    __shared__ float sA[4][16 * DD];
    __shared__ float sK[4][16 * DD];
    const int tid = threadIdx.x, wave = tid >> 5, lane = tid & 31;

    async_copy_16k(sW, Wk, tid);

    const int e0 = blockIdx.x * 64 + wave * 16;
    {   // stage nr = emb[tail] * rel : 2 lanes per edge, 32 dims each (VALU path)
        const int el = lane >> 1, d0 = (lane & 1) * 32;
        int e = e0 + el; if (e >= E) e = E - 1;
        const int tail = eidx[E + e];
        const int rt   = etype[e] - 1;
        const float4* te = (const float4*)(emb + (size_t)tail * DD + d0);
        const float4* re = (const float4*)(relW + (size_t)rt * DD + d0);
        float4* dst = (float4*)(sA[wave] + el * DD + d0);
#pragma unroll
        for (int j = 0; j < 8; ++j) {
            float4 t = te[j], r = re[j];
            float4 o; o.x = t.x * r.x; o.y = t.y * r.y; o.z = t.z * r.z; o.w = t.w * r.w;
            dst[j] = o;
        }
    }
    async_wait_all();
    __syncthreads();
    tile_gemm_16x64(sA[wave], sW, sK[wave], lane);
    __syncthreads();
    // 16 edges * 4 heads = 64 dot products of length 16, 2 per lane
#pragma unroll
    for (int p = 0; p < 2; ++p) {
        const int idx = p * 32 + lane;
        const int el = idx >> 2, h = idx & 3;
        const int e = e0 + el;
        if (e < E) {
            const int head = eidx[e];
            const float* q = Qn + (size_t)head * DD + h * 16;
            const float* k = sK[wave] + el * DD + h * 16;
            float dot = 0.f;
#pragma unroll
            for (int j = 0; j < 16; ++j) dot += q[j] * k[j];
            dot = fminf(10.f, fmaxf(-10.f, dot));
            float ea = expf(dot);
            exp_att[(size_t)e * 4 + h] = ea;
            atomicAdd(&att_norm[(size_t)head * 4 + h], ea);
        }
    }
}

// ------- Kernel 3: per-edge V projection + normalized scatter into kg_emb -----
__global__ void __launch_bounds__(128)
edge_aggv_kernel(const float* __restrict__ emb, const float* __restrict__ relW,
                 const float* __restrict__ Wv,
                 const int* __restrict__ eidx, const int* __restrict__ etype,
                 const float* __restrict__ exp_att, const float* __restrict__ att_norm,
                 float* __restrict__ kg, int E) {
    __shared__ float sW[DD * DD];
    __shared__ float sA[4][16 * DD];
    __shared__ float sV[4][16 * DD];
    const int tid = threadIdx.x, wave = tid >> 5, lane = tid & 31;

    async_copy_16k(sW, Wv, tid);

    const int e0 = blockIdx.x * 64 + wave * 16;
    const int el = lane >> 1, d0 = (lane & 1) * 32;
    {
        int e = e0 + el; if (e >= E) e = E - 1;
        const int tail = eidx[E + e];
        const int rt   = etype[e] - 1;
        const float4* te = (const float4*)(emb + (size_t)tail * DD + d0);
        const float4* re = (const float4*)(relW + (size_t)rt * DD + d0);
        float4* dst = (float4*)(sA[wave] + el * DD + d0);
#pragma unroll
        for (int j = 0; j < 8; ++j) {
            float4 t = te[j], r = re[j];
            float4 o; o.x = t.x * r.x; o.y = t.y * r.y; o.z = t.z * r.z; o.w = t.w * r.w;
            dst[j] = o;
        }
    }
    async_wait_all();
    __syncthreads();
    tile_gemm_16x64(sA[wave], sW, sV[wave], lane);
    __syncthreads();
    const int e = e0 + el;
    if (e < E) {
        const int head = eidx[e];
        const int h0 = d0 >> 4;
        const float a0 = exp_att[(size_t)e * 4 + h0] /
                         (att_norm[(size_t)head * 4 + h0] + 1e-8f);
        const float a1 = exp_att[(size_t)e * 4 + h0 + 1] /
                         (att_norm[(size_t)head * 4 + h0 + 1] + 1e-8f);
#pragma unroll
        for (int j = 0; j < 32; ++j) {
            float a = (j < 16) ? a0 : a1;
            atomicAdd(&kg[(size_t)head * DD + d0 + j],
                      a * sV[wave][el * DD + d0 + j]);
        }
    }
}

// ------- Kernel 4: w = (|kg[h]*rel| * |kg[t]*rel|)^2  + segment max ----------
__global__ void sim_w_kernel(const float* __restrict__ kg, const float* __restrict__ relW,
                             const int* __restrict__ eidx, const int* __restrict__ etype,
                             float* __restrict__ w_e, float* __restrict__ seg_max, int E) {
    int e = blockIdx.x * blockDim.x + threadIdx.x;
    if (e >= E) return;
    const int head = eidx[e], tail = eidx[E + e], rt = etype[e] - 1;
    const float* rel = relW + (size_t)rt * DD;
    const float* hp  = kg + (size_t)head * DD;
    const float* tp  = kg + (size_t)tail * DD;
    float hn2 = 0.f, tn2 = 0.f;
#pragma unroll 8
    for (int d = 0; d < DD; ++d) {
        float a = hp[d] * rel[d]; hn2 += a * a;
        float b = tp[d] * rel[d]; tn2 += b * b;
    }
    float w = hn2 * tn2;                  // (hn*tn)^2 without sqrt
    w_e[e] = w;
    // w >= 0 so IEEE bits order as uints; seg_max zero-initialized
    atomicMax((unsigned int*)&seg_max[head], __float_as_uint(w));
}

// ------- Kernel 5: stable exp + segment sum ----------------------------------
__global__ void seg_exp_kernel(const int* __restrict__ eidx, const float* __restrict__ seg_max,
                               float* __restrict__ w_e, float* __restrict__ seg_sum, int E) {
    int e = blockIdx.x * blockDim.x + threadIdx.x;
    if (e >= E) return;
    const int head = eidx[e];
    float ee = expf(w_e[e] - seg_max[head]);
    w_e[e] = ee;
    atomicAdd(&seg_sum[head], ee);
}

// ------- Kernel 6: agg[head] += a * emb[tail] --------------------------------
__global__ void agg_kernel(const int* __restrict__ eidx, const float* __restrict__ emb,
                           const float* __restrict__ w_e, const float* __restrict__ seg_sum,
                           float* __restrict__ agg, int E) {
    long long idx = (long long)blockIdx.x * blockDim.x + threadIdx.x;
    if (idx >= (long long)E * DD) return;
    const int e = (int)(idx >> 6), d = (int)(idx & 63);
    const int head = eidx[e], tail = eidx[E + e];
    const float a = w_e[e] / (seg_sum[head] + 1e-16f);
    atomicAdd(&agg[(size_t)head * DD + d], a * emb[(size_t)tail * DD + d]);
}

// ------- Kernel 7: L2-normalize + residual accumulate ------------------------
__global__ void finalize_kernel(const float* __restrict__ agg, float* __restrict__ emb,
                                float* __restrict__ res, int N) {
    int n = blockIdx.x * blockDim.x + threadIdx.x;
    if (n >= N) return;
    float s = 0.f;
#pragma unroll 8
    for (int d = 0; d < DD; ++d) { float v = agg[(size_t)n * DD + d]; s += v * v; }
    const float inv = 1.f / fmaxf(sqrtf(s), 1e-12f);
#pragma unroll 8
    for (int d = 0; d < DD; ++d) {
        float v = agg[(size_t)n * DD + d] * inv;
        emb[(size_t)n * DD + d] = v;
        res[(size_t)n * DD + d] += v;
    }
}

extern "C" void kernel_launch(void* const* d_in, const int* in_sizes, int n_in,
                              void* d_out, int out_size, void* d_ws, size_t ws_size,
                              hipStream_t stream) {
    const float* entity = (const float*)d_in[0];
    const float* relW   = (const float*)d_in[1];
    const float* Wq     = (const float*)d_in[2];
    const float* Wk     = (const float*)d_in[3];
    const float* Wv     = (const float*)d_in[4];
    const int*   eidx   = (const int*)d_in[5];
    const int*   etype  = (const int*)d_in[6];
    const int N = in_sizes[0] / DD;
    const int E = in_sizes[6];
    float* out = (float*)d_out;

    // workspace carve-up (floats)
    float* ws = (float*)d_ws;
    size_t off = 0;
    float* emb_cur  = ws + off; off += (size_t)N * DD;
    float* Qn       = ws + off; off += (size_t)N * DD;
    float* kg       = ws + off; off += (size_t)N * DD;
    float* aggb     = ws + off; off += (size_t)N * DD;
    float* att_norm = ws + off; off += (size_t)N * 4;
    float* seg_max  = ws + off; off += (size_t)N;
    float* seg_sum  = ws + off; off += (size_t)N;
    float* exp_att  = ws + off; off += (size_t)E * 4;
    float* w_e      = ws + off; off += (size_t)E;
    (void)ws_size; (void)n_in; (void)out_size;

    hipMemcpyAsync(emb_cur, entity, (size_t)N * DD * sizeof(float),
                   hipMemcpyDeviceToDevice, stream);
    hipMemcpyAsync(out, entity, (size_t)N * DD * sizeof(float),
                   hipMemcpyDeviceToDevice, stream);

    const int nodeBlocks = (N + 63) / 64;
    const int edgeBlocks = (E + 63) / 64;
    const long long totAgg = (long long)E * DD;

    for (int layer = 0; layer < 2; ++layer) {
        hipMemsetAsync(att_norm, 0, (size_t)N * 4 * sizeof(float), stream);
        hipMemsetAsync(kg,       0, (size_t)N * DD * sizeof(float), stream);
        hipMemsetAsync(aggb,     0, (size_t)N * DD * sizeof(float), stream);
        hipMemsetAsync(seg_max,  0, (size_t)N * sizeof(float), stream);
        hipMemsetAsync(seg_sum,  0, (size_t)N * sizeof(float), stream);

        node_gemm_kernel<<<nodeBlocks, 128, 0, stream>>>(
            emb_cur, Wq + (size_t)layer * DD * DD, Qn, N);
        edge_att_kernel<<<edgeBlocks, 128, 0, stream>>>(
            emb_cur, relW, Wk + (size_t)layer * DD * DD, Qn, eidx, etype,
            exp_att, att_norm, E);
        edge_aggv_kernel<<<edgeBlocks, 128, 0, stream>>>(
            emb_cur, relW, Wv + (size_t)layer * DD * DD, eidx, etype,
            exp_att, att_norm, kg, E);
        sim_w_kernel<<<(E + 255) / 256, 256, 0, stream>>>(
            kg, relW, eidx, etype, w_e, seg_max, E);
        seg_exp_kernel<<<(E + 255) / 256, 256, 0, stream>>>(
            eidx, seg_max, w_e, seg_sum, E);
        agg_kernel<<<(int)((totAgg + 255) / 256), 256, 0, stream>>>(
            eidx, emb_cur, w_e, seg_sum, aggb, E);
        finalize_kernel<<<(N + 255) / 256, 256, 0, stream>>>(
            aggb, emb_cur, out, N);
    }
}